// ErrorLoss_23570780520961
// MI455X (gfx1250) — compile-verified
//
#include <hip/hip_runtime.h>

typedef __attribute__((ext_vector_type(2))) float v2f;
typedef __attribute__((ext_vector_type(8))) float v8f;

namespace {
constexpr int ROWS_ = 3;
constexpr int B_ = 16, H_ = 256, W_ = 256;
constexpr int HP_ = H_ - ROWS_ + 1;   // 254
constexpr int WP_ = W_ - ROWS_ + 1;   // 254
constexpr int CP_ = 16;
constexpr int N_ = B_ * HP_ * WP_;    // 1,032,256
constexpr int TILES_ = (N_ + 15) / 16;
constexpr int THREADS_ = 256;
constexpr int NWAVES_ = THREADS_ / 32;
constexpr int BLOCKS_ = 1024;
constexpr int DIF_ELEMS_ = B_ * 2 * H_ * W_;      // 2,097,152 floats (8.4 MB)
constexpr int DIF_VEC4_ = DIF_ELEMS_ / 4;          // 524,288
constexpr size_t DIF_OFFSET_ = 65536;              // ws byte offset for dif buffer
}

__device__ __forceinline__ float block_reduce_sum(float v, float* red) {
  const int tid = threadIdx.x;
  red[tid] = v;
  __syncthreads();
#pragma unroll
  for (int s = THREADS_ / 2; s > 0; s >>= 1) {
    if (tid < s) red[tid] += red[tid + s];
    __syncthreads();
  }
  const float r = red[0];
  __syncthreads();
  return r;
}

// Pass 0: dif = expected - actual_mu  (vectorized, bandwidth-bound)
__global__ __launch_bounds__(THREADS_) void ErrorLoss_diff(
    const float* __restrict__ mu, const float* __restrict__ expd,
    float* __restrict__ dif)
{
  const int i = blockIdx.x * THREADS_ + threadIdx.x;
  if (i < DIF_VEC4_) {
    const float4 e = ((const float4*)expd)[i];
    const float4 m = ((const float4*)mu)[i];
    float4 d;
    d.x = e.x - m.x; d.y = e.y - m.y; d.z = e.z - m.z; d.w = e.w - m.w;
    ((float4*)dif)[i] = d;
  }
}

// Pass 1: fused gather + V_WMMA_F32_16X16X4_F32 projection + reductions.
// 16 rows per wave per tile; 13 K-chunks of 4 cover the 52 features.
// All chosen-index logic is branch-free (v_cndmask), so EXEC is full at WMMA
// and the load stream stays straight-line / clause-batched.
__global__ __launch_bounds__(THREADS_) void ErrorLoss_fused_wmma(
    const float* __restrict__ dif, const float* __restrict__ prun,
    const float* __restrict__ Wm, const float* __restrict__ bias,
    const int* __restrict__ idxc, float* __restrict__ partials)
{
  __shared__ float lds_res[NWAVES_][64];   // 16 rows x 4 cols per wave
  __shared__ float red[THREADS_];

  const int tid  = threadIdx.x;
  const int lane = tid & 31;
  const int wave = tid >> 5;
  const int m    = lane & 15;   // A: row within 16-row tile; B: column index
  const int half = lane >> 4;   // 0 or 1
  const int kb   = half << 1;   // K offset within a 4-chunk: 0 or 2
  const float sv = (kb == 0) ? 1.0f : -1.0f;  // chosen-index fill (1,1,-1,-1)

  // B fragment from W (52x4, row-major): lane holds column m; only m<4 nonzero.
  float bw0[13], bw1[13];
#pragma unroll
  for (int kc = 0; kc < 13; ++kc) {
    const int k0 = kc * 4 + kb;
    bw0[kc] = (m < 4) ? Wm[k0 * 4 + m]       : 0.0f;
    bw1[kc] = (m < 4) ? Wm[(k0 + 1) * 4 + m] : 0.0f;
  }
  const float bias0 = bias[0], bias1 = bias[1], bias2 = bias[2], bias3 = bias[3];

  float acc_w = 0.f, acc_cnt = 0.f, acc_hi = 0.f, acc_lo = 0.f;

  const int gw      = blockIdx.x * NWAVES_ + wave;   // uniform per wave
  const int gstride = gridDim.x * NWAVES_;

  for (int t = gw; t < TILES_; t += gstride) {
    const int n_raw = t * 16 + m;
    const bool valid = n_raw < N_;
    const unsigned n = valid ? (unsigned)n_raw : (unsigned)(N_ - 1);
    const int w = (int)(n % (unsigned)WP_);
    const unsigned t2 = n / (unsigned)WP_;
    const int h = (int)(t2 % (unsigned)HP_);
    const int b = (int)(t2 / (unsigned)HP_);
    const int ic = idxc[n];
    const int prun_base = ((b * CP_) * HP_ + h) * WP_ + w;
    const int dif_base  = ((b * 2) * H_ + h) * W_ + w;

    float piv0 = 0.f, piv1 = 0.f;
    v8f acc = {0.f, 0.f, 0.f, 0.f, 0.f, 0.f, 0.f, 0.f};
#pragma unroll
    for (int kc = 0; kc < 13; ++kc) {
      float a0, a1;
      if (kc < 4) {                       // features 0..15: pruning channels
        const int k0 = kc * 4 + kb;
        a0 = prun[prun_base + k0 * (HP_ * WP_)];
        a1 = prun[prun_base + (k0 + 1) * (HP_ * WP_)];
      } else {                            // features 16..51: mega (d0,d1,d0,d1)
        const int r2 = kc - 4;            // compile-time window index
        const int wi = r2 / 3, wj = r2 % 3;
        const int off0 = dif_base + wi * W_ + wj;
        const float d0 = dif[off0];                // channel 0
        const float d1 = dif[off0 + H_ * W_];      // channel 1
        const bool chosen = (r2 == ic);
        a0 = chosen ? sv : d0;                     // branch-free select
        a1 = chosen ? sv : d1;
        piv0 = chosen ? d0 : piv0;                 // pivot by selection,
        piv1 = chosen ? d1 : piv1;                 // no extra loads
      }
      const v2f av = {a0, a1};
      const v2f bv = {bw0[kc], bw1[kc]};
      acc = __builtin_amdgcn_wmma_f32_16x16x4_f32(false, av, false, bv,
                                                  (short)0, acc, false, false);
    }

    // D layout: lane, vgpr v -> (row = v + 8*half, col = lane&15). Spill cols 0..3.
    if (m < 4) {
#pragma unroll
      for (int v = 0; v < 8; ++v)
        lds_res[wave][(v + half * 8) * 4 + m] = acc[v];
    }
    __builtin_amdgcn_wave_barrier();

    if (lane < 16 && valid) {
      const float r0  = lds_res[wave][lane * 4 + 0] + bias0;
      const float r1  = lds_res[wave][lane * 4 + 1] + bias1;
      const float r2v = lds_res[wave][lane * 4 + 2] + bias2;
      const float r3  = lds_res[wave][lane * 4 + 3] + bias3;
      acc_w += fabsf(r2v - r0) + fabsf(r3 - r1);
      const bool inside = (piv0 - r0 >= 0.f) && (piv1 - r1 >= 0.f) &&
                          (piv0 - r2v <= 0.f) && (piv1 - r3 <= 0.f);
      acc_cnt += inside ? 1.0f : 0.0f;
      const float h0 = fmaxf(piv0 - r2v, 0.f), h1 = fmaxf(piv1 - r3, 0.f);
      const float l0 = fmaxf(r0 - piv0, 0.f),  l1 = fmaxf(r1 - piv1, 0.f);
      acc_hi += h0 * h0 + h1 * h1;
      acc_lo += l0 * l0 + l1 * l1;
    }
    __builtin_amdgcn_wave_barrier();
  }

  float r;
  r = block_reduce_sum(acc_w,   red); if (tid == 0) partials[blockIdx.x * 4 + 0] = r;
  r = block_reduce_sum(acc_cnt, red); if (tid == 0) partials[blockIdx.x * 4 + 1] = r;
  r = block_reduce_sum(acc_hi,  red); if (tid == 0) partials[blockIdx.x * 4 + 2] = r;
  r = block_reduce_sum(acc_lo,  red); if (tid == 0) partials[blockIdx.x * 4 + 3] = r;
}

__global__ __launch_bounds__(THREADS_) void ErrorLoss_finalize(
    const float* __restrict__ partials, float* __restrict__ out)
{
  __shared__ float red[THREADS_];
  float sw = 0.f, sc = 0.f, shi = 0.f, slo = 0.f;
  for (int i = threadIdx.x; i < BLOCKS_; i += THREADS_) {
    sw  += partials[i * 4 + 0];
    sc  += partials[i * 4 + 1];
    shi += partials[i * 4 + 2];
    slo += partials[i * 4 + 3];
  }
  sw  = block_reduce_sum(sw,  red);
  sc  = block_reduce_sum(sc,  red);
  shi = block_reduce_sum(shi, red);
  slo = block_reduce_sum(slo, red);
  if (threadIdx.x == 0) {
    const float invN   = 1.0f / (float)N_;
    const float p_in   = sc * invN;
    const float thresh = __expf(__logf(0.9f) / (float)(H_ * W_)); // 0.9^(1/65536)
    const float penalty = (shi + slo) * (0.5f * invN);             // HAMMER = 1
    float loss = 0.25f * sw;                                       // LORRIS
    if (p_in < thresh) loss += penalty;
    out[0] = loss;
  }
}

extern "C" void kernel_launch(void* const* d_in, const int* in_sizes, int n_in,
                              void* d_out, int out_size, void* d_ws, size_t ws_size,
                              hipStream_t stream) {
  (void)in_sizes; (void)n_in; (void)out_size; (void)ws_size;
  const float* mu    = (const float*)d_in[0];   // actual_mu      [16,2,256,256]
  const float* prun  = (const float*)d_in[1];   // actual_pruning [16,16,254,254]
  const float* expd  = (const float*)d_in[2];   // expected       [16,2,256,256]
  const float* Wm    = (const float*)d_in[3];   // W [52,4]
  const float* bias  = (const float*)d_in[4];   // b [4]
  const int*   idxc  = (const int*)d_in[5];     // index_choice [N]

  float* partials = (float*)d_ws;                                // 16 KB
  float* dif      = (float*)((char*)d_ws + DIF_OFFSET_);         // 8.4 MB

  ErrorLoss_diff<<<(DIF_VEC4_ + THREADS_ - 1) / THREADS_, THREADS_, 0, stream>>>(
      mu, expd, dif);
  ErrorLoss_fused_wmma<<<BLOCKS_, THREADS_, 0, stream>>>(
      dif, prun, Wm, bias, idxc, partials);
  ErrorLoss_finalize<<<1, THREADS_, 0, stream>>>(partials, (float*)d_out);
}